// WaveNetworkEnhanced_73186242724048
// MI455X (gfx1250) — compile-verified
//
#include <hip/hip_runtime.h>
#include <hip/hip_bf16.h>
#include <math.h>

#define Hd      512
#define NHEAD   4
#define DHEAD   128
#define BATCH   2
#define SEQ     256
#define MROWS   (BATCH*SEQ)     // 512
#define PI_F    3.14159265358979f

typedef __attribute__((ext_vector_type(2))) float v2f;
typedef __attribute__((ext_vector_type(8))) float v8f;

__device__ __forceinline__ float sigm_(float x)  { return 1.0f / (1.0f + __expf(-x)); }
__device__ __forceinline__ float tanh_(float x)  { return 1.0f - 2.0f / (__expf(2.0f * x) + 1.0f); }

// ---------------------------------------------------------------------------
// Generic f32 WMMA GEMM:  C[M,N] (+)= alpha * (Ashift[M,K] @ B[K,N]) + alpha*bias
//   A row-major, lda=K.  B element [k,n] = Bm[k*bRowStride + n*bColStride].
//   timeShift/seqLen: A row m reads row (m+timeShift) when the shifted time
//   index stays inside [0,seqLen) within its batch; otherwise the row
//   contributes zeros (conv zero-padding) via a clamped in-bounds row plus a
//   0/1 float mask on the A fragment (branch-free inner loop).
//   alpha = softmax(scaleW)[scaleIdx] when scaleW != nullptr, else 1.
// One wave = one 16x16 C tile via V_WMMA_F32_16X16X4_F32; block = 4 waves.
// Inner loop is software-pipelined: fragments for step k+1 are loaded before
// the WMMA of step k, so the WMMA never waits on just-issued loads.
// ---------------------------------------------------------------------------
template <int K>
__global__ void gemm_wmma_f32(const float* __restrict__ A,
                              const float* __restrict__ Bm,
                              const float* __restrict__ bias,
                              const float* __restrict__ scaleW, int scaleIdx,
                              float* __restrict__ C,
                              int M, int N,
                              int bRowStride, int bColStride,
                              int timeShift, int seqLen, int accum)
{
    const int lane  = threadIdx.x & 31;
    const int wid   = threadIdx.x >> 5;
    const int tileM = blockIdx.y * 32 + (wid >> 1) * 16;
    const int tileN = blockIdx.x * 32 + (wid & 1) * 16;
    const int hh    = lane >> 4;      // lane half: K sub-block
    const int l15   = lane & 15;

    // A-fragment row for this lane (rows replicated across the two lane halves)
    const int m = tileM + l15;
    int   rowA  = m;
    float amask = 1.0f;
    if (seqLen > 0) {
        const int t  = m % seqLen;
        const int tt = t + timeShift;
        if (tt >= 0 && tt < seqLen) rowA = m + timeShift;   // valid shifted row
        else                        amask = 0.0f;           // zero-pad: load own row, mask it
    }
    const int n = tileN + l15;

    const float* ap  = A  + (size_t)rowA * K + 2 * hh;      // float2 per step
    const float* bp0 = Bm + (size_t)n * bColStride + (size_t)(2 * hh) * bRowStride;
    const float* bp1 = bp0 + bRowStride;
    const size_t bstep = (size_t)4 * bRowStride;

    // prologue: fetch fragments for step 0
    float2 av = *(const float2*)ap;
    float  bx = *bp0;
    float  by = *bp1;

    v8f acc = {0.f, 0.f, 0.f, 0.f, 0.f, 0.f, 0.f, 0.f};
    constexpr int STEPS = K / 4;
#pragma unroll 8
    for (int s = 0; s < STEPS - 1; ++s) {
        // issue loads for step s+1 BEFORE the WMMA of step s
        ap  += 4;
        bp0 += bstep;
        bp1 += bstep;
        const float2 avn = *(const float2*)ap;
        const float  bxn = *bp0;
        const float  byn = *bp1;

        v2f a; a.x = av.x * amask; a.y = av.y * amask;
        v2f b; b.x = bx;           b.y = by;
        acc = __builtin_amdgcn_wmma_f32_16x16x4_f32(false, a, false, b,
                                                    (short)0, acc, false, false);
        av = avn; bx = bxn; by = byn;
    }
    {   // epilogue: last K-step
        v2f a; a.x = av.x * amask; a.y = av.y * amask;
        v2f b; b.x = bx;           b.y = by;
        acc = __builtin_amdgcn_wmma_f32_16x16x4_f32(false, a, false, b,
                                                    (short)0, acc, false, false);
    }

    float alpha = 1.0f;
    if (scaleW) {
        const float e0 = __expf(scaleW[0]), e1 = __expf(scaleW[1]), e2 = __expf(scaleW[2]);
        const float inv = 1.0f / (e0 + e1 + e2);
        alpha = (scaleIdx == 0 ? e0 : (scaleIdx == 1 ? e1 : e2)) * inv;
    }
    const float badd = bias ? alpha * bias[n] : 0.0f;
#pragma unroll
    for (int r = 0; r < 8; ++r) {
        const int mr = tileM + hh * 8 + r;        // C: VGPR r -> row r (+8 for hi half)
        const size_t idx = (size_t)mr * N + n;
        float v = alpha * acc[r] + badd;
        if (accum) v += C[idx];
        C[idx] = v;
    }
}

// ---------------------------------------------------------------------------
// Wave attention: per (b,h,i) block, 128 threads (one per d).
// scores[i,j,d] = amp[d]*cos(omega[d]*(i-j)+ph[d]) / sqrt(D); softmax over j;
// ctx[i,d] = sum_j softmax * v[j,d].  Two-pass softmax, v staged in LDS chunks.
// ---------------------------------------------------------------------------
#define VCHUNK 64
__global__ void wave_attention(const float* __restrict__ Q,
                               const float* __restrict__ Kb,
                               const float* __restrict__ V,
                               float* __restrict__ CTX)
{
    __shared__ __align__(16) float vtile[VCHUNK * DHEAD];   // 32 KB
    const int bid = blockIdx.x;
    const int i = bid % SEQ;
    const int h = (bid / SEQ) % NHEAD;
    const int b = bid / (SEQ * NHEAD);
    const int d = threadIdx.x;                               // 0..127

    const size_t rowBase = ((size_t)(b * SEQ + i)) * Hd + h * DHEAD + d;
    const float qv = Q[rowBase];
    const float kv = Kb[rowBase];
    const float vv = V[rowBase];
    const float omega = PI_F * sigm_(qv);
    const float amp   = sigm_(kv);
    const float ph    = PI_F * tanh_(vv);
    const float scl   = 0.08838834764831845f;   // 1/sqrt(128)

    // pass 1: row max over j (pure transcendental, no memory)
    float mmax = -3.4e38f;
#pragma unroll 4
    for (int j = 0; j < SEQ; ++j) {
        float s = amp * __cosf(omega * (float)(i - j) + ph) * scl;
        mmax = fmaxf(mmax, s);
    }

    // pass 2: exp-sum and ctx, v chunks through LDS
    const float* vsrc = V + ((size_t)b * SEQ) * Hd + h * DHEAD;
    float sum = 0.0f, ctx = 0.0f;
    for (int j0 = 0; j0 < SEQ; j0 += VCHUNK) {
        __syncthreads();
        for (int r = 0; r < VCHUNK; r += 4) {                 // 4 rows / iter, float4
            int j = r + (threadIdx.x >> 5);
            int c = (threadIdx.x & 31) * 4;
            *(float4*)&vtile[j * DHEAD + c] =
                *(const float4*)&vsrc[(size_t)(j0 + j) * Hd + c];
        }
        __syncthreads();
#pragma unroll 4
        for (int jj = 0; jj < VCHUNK; ++jj) {
            int j = j0 + jj;
            float s = amp * __cosf(omega * (float)(i - j) + ph) * scl;
            float e = __expf(s - mmax);
            sum += e;
            ctx += e * vtile[jj * DHEAD + d];
        }
    }
    CTX[rowBase] = ctx / sum;                                  // (B,S,NH*D) layout
}

// ---------------------------------------------------------------------------
// Phase branch: mean over S of cos(p), sin(p) per (b, channel).
// cos p = re * rsqrt(re^2+im^2), sin p = im * rsqrt(...).
// ---------------------------------------------------------------------------
__global__ void phase_reduce(const float* __restrict__ CR,
                             float* __restrict__ MC, float* __restrict__ MS)
{
    const int c = (blockIdx.x % (Hd / 256)) * 256 + threadIdx.x;
    const int b = blockIdx.x / (Hd / 256);
    const float* base = CR + (size_t)b * SEQ * 2 * Hd;
    float sc = 0.0f, ss = 0.0f;
    for (int s = 0; s < SEQ; ++s) {
        float re = base[(size_t)s * 2 * Hd + c];
        float im = base[(size_t)s * 2 * Hd + Hd + c];
        float inv = rsqrtf(re * re + im * im + 1e-20f);
        sc += re * inv;
        ss += im * inv;
    }
    MC[b * Hd + c] = sc * (1.0f / SEQ);
    MS[b * Hd + c] = ss * (1.0f / SEQ);
}

// new_p = p + 0.1*(cos(p+shift)*mc + sin(p+shift)*ms);  nc = mag*[cos,sin](new_p)
// computed with angle-addition identities (no atan2 needed).
__global__ void phase_apply(const float* __restrict__ CR,
                            const float* __restrict__ shifts,
                            const float* __restrict__ MC,
                            const float* __restrict__ MS,
                            float* __restrict__ NC)
{
    const int idx = blockIdx.x * blockDim.x + threadIdx.x;   // over B*S*H
    const int c  = idx % Hd;
    const int bs = idx / Hd;
    const int b  = bs / SEQ;
    const size_t o = (size_t)bs * 2 * Hd + c;
    const float re = CR[o], im = CR[o + Hd];
    const float mag = sqrtf(re * re + im * im + 1e-5f);
    const float inv = rsqrtf(re * re + im * im + 1e-20f);
    const float cp = re * inv, sp = im * inv;
    const float sh = shifts[c];
    const float csh = __cosf(sh), ssh = __sinf(sh);
    const float mc = MC[b * Hd + c], ms = MS[b * Hd + c];
    const float mixed = (cp * csh - sp * ssh) * mc + (sp * csh + cp * ssh) * ms;
    const float dp = 0.1f * mixed;
    const float cd = __cosf(dp), sd = __sinf(dp);
    NC[o]      = mag * (cp * cd - sp * sd);
    NC[o + Hd] = mag * (sp * cd + cp * sd);
}

// ---------------------------------------------------------------------------
// out = LayerNorm(x + combined) * g + b   — one block per (b,s) row, 256 thr.
// ---------------------------------------------------------------------------
__global__ void add_layernorm(const float* __restrict__ x,
                              const float* __restrict__ comb,
                              const float* __restrict__ g,
                              const float* __restrict__ bta,
                              float* __restrict__ out)
{
    __shared__ float red[256];
    const int row = blockIdx.x;
    const int t = threadIdx.x;
    const size_t base = (size_t)row * Hd;
    float v0 = x[base + t]       + comb[base + t];
    float v1 = x[base + 256 + t] + comb[base + 256 + t];
    red[t] = v0 + v1;
    __syncthreads();
    for (int off = 128; off > 0; off >>= 1) {
        if (t < off) red[t] += red[t + off];
        __syncthreads();
    }
    const float mu = red[0] * (1.0f / Hd);
    __syncthreads();
    const float d0 = v0 - mu, d1 = v1 - mu;
    red[t] = d0 * d0 + d1 * d1;
    __syncthreads();
    for (int off = 128; off > 0; off >>= 1) {
        if (t < off) red[t] += red[t + off];
        __syncthreads();
    }
    const float inv = rsqrtf(red[0] * (1.0f / Hd) + 1e-5f);
    out[base + t]       = d0 * inv * g[t]       + bta[t];
    out[base + 256 + t] = d1 * inv * g[256 + t] + bta[256 + t];
}

// ---------------------------------------------------------------------------
extern "C" void kernel_launch(void* const* d_in, const int* in_sizes, int n_in,
                              void* d_out, int out_size, void* d_ws, size_t ws_size,
                              hipStream_t stream)
{
    const float* x      = (const float*)d_in[0];
    const float* Wq     = (const float*)d_in[1];
    const float* bq     = (const float*)d_in[2];
    const float* Wk     = (const float*)d_in[3];
    const float* bk     = (const float*)d_in[4];
    const float* Wv     = (const float*)d_in[5];
    const float* bv     = (const float*)d_in[6];
    const float* Wo     = (const float*)d_in[7];
    const float* bo     = (const float*)d_in[8];
    const float* cw1    = (const float*)d_in[9];
    const float* cb1    = (const float*)d_in[10];
    const float* cw2    = (const float*)d_in[11];
    const float* cb2    = (const float*)d_in[12];
    const float* cw3    = (const float*)d_in[13];
    const float* cb3    = (const float*)d_in[14];
    const float* scalew = (const float*)d_in[15];
    const float* Wwl    = (const float*)d_in[16];
    const float* bwl    = (const float*)d_in[17];
    const float* Wtc    = (const float*)d_in[18];
    const float* btc    = (const float*)d_in[19];
    const float* Wfc    = (const float*)d_in[20];
    const float* bfc    = (const float*)d_in[21];
    const float* shifts = (const float*)d_in[22];
    const float* Wproj  = (const float*)d_in[23];
    const float* bproj  = (const float*)d_in[24];
    const float* ln_g   = (const float*)d_in[25];
    const float* ln_b   = (const float*)d_in[26];

    // workspace carve-out (floats); total ~13.6 MB
    float* ws   = (float*)d_ws;
    float* Qb   = ws;                 // 512*512
    float* Kb   = Qb   + 262144;
    float* Vb   = Kb   + 262144;
    float* CTX  = Vb   + 262144;
    float* CR   = CTX  + 262144;      // 512*1024
    float* WAVE = CR   + 524288;
    float* COMB = WAVE + 262144;
    float* WVLT = COMB + 262144;
    float* NC   = WVLT + 262144;      // 512*1024
    float* PHS  = NC   + 524288;
    float* CMB  = PHS  + 262144;
    float* MC   = CMB  + 262144;      // 1024
    float* MS   = MC   + 1024;

    const dim3 blk(128);
    const dim3 gN512(512 / 32, MROWS / 32);
    const dim3 gN1024(1024 / 32, MROWS / 32);

    // q/k/v projections + complex transform (x @ Wtc)
    gemm_wmma_f32<512><<<gN512,  blk, 0, stream>>>(x, Wq,  bq,  nullptr, 0, Qb, MROWS, 512,  512,  1, 0, 0, 0);
    gemm_wmma_f32<512><<<gN512,  blk, 0, stream>>>(x, Wk,  bk,  nullptr, 0, Kb, MROWS, 512,  512,  1, 0, 0, 0);
    gemm_wmma_f32<512><<<gN512,  blk, 0, stream>>>(x, Wv,  bv,  nullptr, 0, Vb, MROWS, 512,  512,  1, 0, 0, 0);
    gemm_wmma_f32<512><<<gN1024, blk, 0, stream>>>(x, Wtc, btc, nullptr, 0, CR, MROWS, 1024, 1024, 1, 0, 0, 0);

    // wave attention -> ctx in (B,S,H) layout, then output projection
    wave_attention<<<BATCH * NHEAD * SEQ, 128, 0, stream>>>(Qb, Kb, Vb, CTX);
    gemm_wmma_f32<512><<<gN512, blk, 0, stream>>>(CTX, Wo, bo, nullptr, 0, WAVE, MROWS, 512, 512, 1, 0, 0, 0);

    // dilated conv bank as shifted GEMMs accumulated into COMB (alpha = softmax(scale_w)[i])
    // conv1: K=3 pad=1 ; conv2: K=5 pad=2 ; conv3: K=9 pad=4
    for (int t = 0; t < 3; ++t)
        gemm_wmma_f32<512><<<gN512, blk, 0, stream>>>(x, cw1 + t, (t == 0) ? cb1 : nullptr,
                                                      scalew, 0, COMB, MROWS, 512,
                                                      3, 512 * 3, t - 1, SEQ, (t == 0) ? 0 : 1);
    for (int t = 0; t < 5; ++t)
        gemm_wmma_f32<512><<<gN512, blk, 0, stream>>>(x, cw2 + t, (t == 0) ? cb2 : nullptr,
                                                      scalew, 1, COMB, MROWS, 512,
                                                      5, 512 * 5, t - 2, SEQ, 1);
    for (int t = 0; t < 9; ++t)
        gemm_wmma_f32<512><<<gN512, blk, 0, stream>>>(x, cw3 + t, (t == 0) ? cb3 : nullptr,
                                                      scalew, 2, COMB, MROWS, 512,
                                                      9, 512 * 9, t - 4, SEQ, 1);
    gemm_wmma_f32<512><<<gN512, blk, 0, stream>>>(COMB, Wwl, bwl, nullptr, 0, WVLT, MROWS, 512, 512, 1, 0, 0, 0);

    // phase branch
    phase_reduce<<<BATCH * (Hd / 256), 256, 0, stream>>>(CR, MC, MS);
    phase_apply<<<(BATCH * SEQ * Hd) / 256, 256, 0, stream>>>(CR, shifts, MC, MS, NC);
    gemm_wmma_f32<1024><<<gN512, blk, 0, stream>>>(NC, Wfc, bfc, nullptr, 0, PHS, MROWS, 512, 512, 1, 0, 0, 0);

    // combined = [wave|wavelet|phase] @ Wproj + bproj, as three accumulating chunks
    gemm_wmma_f32<512><<<gN512, blk, 0, stream>>>(WAVE, Wproj,                 bproj,   nullptr, 0, CMB, MROWS, 512, 512, 1, 0, 0, 0);
    gemm_wmma_f32<512><<<gN512, blk, 0, stream>>>(WVLT, Wproj + 512 * 512,     nullptr, nullptr, 0, CMB, MROWS, 512, 512, 1, 0, 0, 1);
    gemm_wmma_f32<512><<<gN512, blk, 0, stream>>>(PHS,  Wproj + 2 * 512 * 512, nullptr, nullptr, 0, CMB, MROWS, 512, 512, 1, 0, 0, 1);

    // residual + layernorm
    add_layernorm<<<MROWS, 256, 0, stream>>>(x, CMB, ln_g, ln_b, (float*)d_out);
}